// MXQuantizerSTE_75806172774747
// MI455X (gfx1250) — compile-verified
//
#include <hip/hip_runtime.h>

// MXFP8-E4M3 (block=32, E8M0 shared scale) quantize->dequantize, fp32 in/out.
// Streaming, bandwidth-bound: non-temporal 128-bit global loads/stores,
// wave32-native per-block amax via ds_swizzle xor exchanges (integer-domain
// max: |x| bit patterns order like floats and NaN auto-poisons the block),
// element quantization via the gfx1250 FP8 hardware converters when available.

typedef float v4f __attribute__((ext_vector_type(4)));  // clang vector: NT-able

#define HAVE_HW_FP8 (__has_builtin(__builtin_amdgcn_cvt_pk_fp8_f32) && \
                     __has_builtin(__builtin_amdgcn_cvt_pk_f32_fp8))

// Single ds_swizzle_b32 xor-exchange + unsigned max. Group-of-32 encoding:
// offset[15]=0, xor_mask=offset[14:10], or_mask=offset[9:5]=0, and_mask=0x1F.
template <int IMM>
__device__ __forceinline__ unsigned swz_xor_maxu(unsigned m) {
    unsigned r = (unsigned)__builtin_amdgcn_ds_swizzle((int)m, IMM);
    return (m > r) ? m : r;
}

#if !HAVE_HW_FP8
// Fallback: quantize one normalized value onto the E4M3 grid (subnormals,
// emin=-6, 3 mantissa bits), round-to-nearest-even via v_rndne_f32.
// sh = 3 - max(floor(log2|nv|), -6) = min(130 - biased_exp, 9).
__device__ __forceinline__ float qdq_e4m3(float nv) {
    int e  = (int)((__float_as_uint(nv) >> 23) & 0xFFu);
    int sh = 130 - e;
    sh = (sh < 9) ? sh : 9;
    float q = __builtin_ldexpf(__builtin_rintf(__builtin_ldexpf(nv, sh)), -sh);
    return fminf(fmaxf(q, -240.0f), 240.0f);
}
#endif

__global__ __launch_bounds__(256) void mxfp8_qdq_kernel(
    const v4f* __restrict__ in, v4f* __restrict__ out, int n4) {
    const int idx = blockIdx.x * 256 + threadIdx.x;
    if (idx >= n4) return;

    // Run-ahead prefetch on the streaming input (global_prefetch_b8).
    __builtin_prefetch((const char*)(in + idx) + 16384, 0, 3);

    // Dead-on-arrival data: non-temporal load (evict-first, no L2 rinse).
    const v4f v = __builtin_nontemporal_load(&in[idx]);

    // amax in the unsigned-bit domain: for non-negative floats, bit-pattern
    // order == value order, and NaN payloads (> 0x7F800000) dominate, which
    // reproduces jnp.max's NaN propagation -> block invalidation.
    unsigned b0 = __float_as_uint(v.x) & 0x7FFFFFFFu;
    unsigned b1 = __float_as_uint(v.y) & 0x7FFFFFFFu;
    unsigned b2 = __float_as_uint(v.z) & 0x7FFFFFFFu;
    unsigned b3 = __float_as_uint(v.w) & 0x7FFFFFFFu;
    unsigned mb = (b0 > b1) ? b0 : b1;
    unsigned mc = (b2 > b3) ? b2 : b3;
    mb = (mb > mc) ? mb : mc;

    // One MX block (32 elems) == 8 lanes x float4; waves are 128-element
    // aligned so blocks coincide with 8-lane groups.
    mb = swz_xor_maxu<0x041F>(mb);   // lane ^ 1
    mb = swz_xor_maxu<0x081F>(mb);   // lane ^ 2
    mb = swz_xor_maxu<0x101F>(mb);   // lane ^ 4

    // Shared scale 2^(floor(log2 amax) - 3) with E8M0 no-subnormal flush:
    //   be = biased exponent of amax; sE = be - 130.
    //   valid: amax > 0, finite, sE >= -127  <=>  3 <= be <= 254.
    //   sE == -127 rounds up to 2^-126 (chop doubling case): t = max(be-3, 1).
    const unsigned be = mb >> 23;                 // sign already masked off
    const bool valid = (be >= 3u) && (be <= 254u);
    const unsigned t = (be >= 4u) ? (be - 3u) : 1u;
    const float scale = valid ? __uint_as_float(t << 23)          : 0.0f;
    const float inv   = valid ? __uint_as_float((254u - t) << 23) : 0.0f;

    // Normalized values are < 16, so the OCP E4M3 hardware grid is
    // bit-identical to the reference (4,3) grid and saturation never fires.
    const float n0 = v.x * inv, n1 = v.y * inv;
    const float n2 = v.z * inv, n3 = v.w * inv;

    float q0, q1, q2, q3;
#if HAVE_HW_FP8
    const int p01 = __builtin_amdgcn_cvt_pk_fp8_f32(n0, n1, 0, false);
    const int p23 = __builtin_amdgcn_cvt_pk_fp8_f32(n2, n3, 0, false);
    const auto f01 = __builtin_amdgcn_cvt_pk_f32_fp8(p01, false);
    const auto f23 = __builtin_amdgcn_cvt_pk_f32_fp8(p23, false);
    q0 = f01[0]; q1 = f01[1]; q2 = f23[0]; q3 = f23[1];
#else
    q0 = qdq_e4m3(n0); q1 = qdq_e4m3(n1);
    q2 = qdq_e4m3(n2); q3 = qdq_e4m3(n3);
#endif

    v4f r;
    r.x = valid ? q0 * scale : 0.0f;
    r.y = valid ? q1 * scale : 0.0f;
    r.z = valid ? q2 * scale : 0.0f;
    r.w = valid ? q3 * scale : 0.0f;

    __builtin_nontemporal_store(r, &out[idx]);
}

extern "C" void kernel_launch(void* const* d_in, const int* in_sizes, int n_in,
                              void* d_out, int out_size, void* d_ws, size_t ws_size,
                              hipStream_t stream) {
    (void)n_in; (void)out_size; (void)d_ws; (void)ws_size;
    const float* x = (const float*)d_in[0];
    float* out = (float*)d_out;

    const long long n = (long long)in_sizes[0];   // 4096*8192, divisible by 128
    const int n4 = (int)(n / 4);                  // float4 elements
    const int blocks = (n4 + 255) / 256;

    mxfp8_qdq_kernel<<<blocks, 256, 0, stream>>>(
        (const v4f*)x, (v4f*)out, n4);
}